// InterAgentSoftPenalty_85607288144429
// MI455X (gfx1250) — compile-verified
//
#include <hip/hip_runtime.h>

// InterAgentSoftPenalty on gfx1250 (MI455X, wave32).
//
// d2[i,j] = |pi|^2 + |pj|^2 - 2 pi.pj  is a rank-2 update -> use
// V_WMMA_F32_16X16X4_F32 with K=4 packing:
//   A row i = [2s*xi, 2s*yi, -s, -s*(xi^2+yi^2)]
//   B col j = [xj,    yj,    xj^2+yj^2,      1 ]
//   A.B = -s * d2[i,j],  s = log2(e)/sigma^2
// so the accumulator already holds the exp2 argument; pen = exp2(acc).

typedef __attribute__((ext_vector_type(2))) float v2f;
typedef __attribute__((ext_vector_type(8))) float v8f;

#define NAGENTS 64
#define WEIGHT_F 10.0f
// log2(e) / sigma^2 = 1.4426950408889634 / 0.01
#define SCALE_F 144.26950408889634f

__global__ __launch_bounds__(256)
void InterAgentSoftPenalty_kernel(const float* __restrict__ states,
                                  float* __restrict__ out) {
    const int lane = threadIdx.x & 31;
    const int wave = threadIdx.x >> 5;
    const int wid  = blockIdx.x * 8 + wave;   // one wave per (b,h); grid exact

    const float* base = states + (size_t)wid * (4 * NAGENTS);

    const int m    = lane & 15;   // row (A) / col (B) within 16-group
    const int half = lane >> 4;   // K-half: lanes 0-15 -> K=0,1; 16-31 -> K=2,3

    // Build A fragments (16x4) and B fragments (4x16) for the four 16-groups.
    v2f afrag[4];
    v2f bfrag[4];
#pragma unroll
    for (int g = 0; g < 4; ++g) {
        const int idx = g * 16 + m;
        const float2 p = *(const float2*)(base + 4 * idx); // (x,y), 8B aligned
        const float x = p.x;
        const float y = p.y;
        const float r2 = x * x + y * y;
        // A: 16x4 f32 layout -> lane half 0: V0=K0, V1=K1; half 1: V0=K2, V1=K3
        afrag[g].x = half ? (-SCALE_F)       : (2.0f * SCALE_F * x);
        afrag[g].y = half ? (-SCALE_F * r2)  : (2.0f * SCALE_F * y);
        // B: 4x16 f32 layout (rows striped over lanes within a VGPR)
        bfrag[g].x = half ? r2   : x;
        bfrag[g].y = half ? 1.0f : y;
    }

    // Only tiles with ti >= tj contribute (strict lower triangle i > j).
    float sum = 0.0f;
#pragma unroll
    for (int ti = 0; ti < 4; ++ti) {
#pragma unroll
        for (int tj = 0; tj <= ti; ++tj) {
            v8f c = {};
            // 8 args: (neg_a, A, neg_b, B, c_mod, C, reuse_a, reuse_b)
            c = __builtin_amdgcn_wmma_f32_16x16x4_f32(
                    false, afrag[ti], false, bfrag[tj],
                    (short)0, c, false, false);
#pragma unroll
            for (int r = 0; r < 8; ++r) {
                float e = __builtin_amdgcn_exp2f(c[r]);
                if (ti == tj) {
                    // C/D layout: row = r + 8*half, col = lane&15.
                    const int row = r + 8 * half;
                    if (row <= m) e = 0.0f;   // keep strictly lower (i > j)
                }
                sum += e;
            }
        }
    }

    // wave32 reduction
#pragma unroll
    for (int off = 16; off >= 1; off >>= 1)
        sum += __shfl_xor(sum, off, 32);

    if (lane == 0)
        out[wid] = WEIGHT_F * sum;
}

extern "C" void kernel_launch(void* const* d_in, const int* in_sizes, int n_in,
                              void* d_out, int out_size, void* d_ws, size_t ws_size,
                              hipStream_t stream) {
    (void)in_sizes; (void)n_in; (void)d_ws; (void)ws_size;
    const float* states = (const float*)d_in[0];
    float* out = (float*)d_out;
    // out_size = 8192 (b,h) slices; one wave32 each; 8 waves per 256-thread block.
    const int nSlices = out_size;           // 8192
    const int blocks  = nSlices / 8;        // 1024, exact
    InterAgentSoftPenalty_kernel<<<blocks, 256, 0, stream>>>(states, out);
}